// Contextual_Patches_Reconstruction_Module_66159676228282
// MI455X (gfx1250) — compile-verified
//
#include <hip/hip_runtime.h>
#include <math.h>

// ---------------------------------------------------------------------------
// Contextual patch attention, MI455X (gfx1250, wave32, WMMA).
//   b   : (4,64,80,80) f32       mask : (1,1,80,80) f32
//   out : (4,64,80,80) f32
// Decomposition: S = diag(10*mm*inv_norm) * (xf xf^T); yi = colsoftmax(S)*mm;
//                patches = yi^T raw_w; overlap-add.
// FP32 WMMA (V_WMMA_F32_16X16X4_F32): SOFTMAX_SCALE=10 with logit magnitude
// up to ~240 rules out bf16/fp8 matrix paths.
// GEMM1 exploits Gram-matrix symmetry (upper-triangular tile blocks only).
// ---------------------------------------------------------------------------

typedef __attribute__((ext_vector_type(2))) float v2f;
typedef __attribute__((ext_vector_type(8))) float v8f;

#define NB   4
#define NC   64
#define H    80
#define W    80
#define HS   40          // half-res grid
#define L    1600        // HS*HS patch positions
#define DQ   576         // C*3*3
#define DV   1024        // C*4*4

// workspace layout (float offsets)
#define XF_OFF    ((size_t)0)                       // NB*L*DQ   = 3,686,400
#define SCALE_OFF (XF_OFF + (size_t)NB*L*DQ)        // NB*L      = 6,400
#define MM_OFF    (SCALE_OFF + (size_t)NB*L)        // L         = 1,600
#define RW_OFF    (MM_OFF + (size_t)L)              // NB*L*DV   = 6,553,600
#define S_OFF     (RW_OFF + (size_t)NB*L*DV)        // NB*L*L    = 10,240,000
#define P_OFF     (S_OFF + (size_t)NB*L*L)          // NB*L*DV   = 6,553,600
// total ~27.04M floats = ~103 MiB

// ---------------------------------------------------------------------------
// 1) Build xf (3x3 patches of half-res map), raw_w (4x4 patches of full map),
//    mask term mm[l], fused row scale 10*mm/max(||xf||,1e-4).
// ---------------------------------------------------------------------------
__global__ void prep_kernel(const float* __restrict__ src,
                            const float* __restrict__ mask,
                            float* __restrict__ xf, float* __restrict__ rw,
                            float* __restrict__ scale, float* __restrict__ mm) {
  const int l = blockIdx.x;      // 0..1599
  const int b = blockIdx.y;      // 0..3
  const int t = threadIdx.x;     // 0..63
  const int hs = l / HS, ws = l % HS;
  const float* sb = src + (size_t)b * NC * H * W;

  float sumsq = 0.f;
  for (int d = t; d < DQ; d += 64) {
    const int c = d / 9, r = d % 9, u = r / 3, v = r % 3;
    const int y = 2 * (hs + u - 1), x = 2 * (ws + v - 1);
    float val = 0.f;
    if ((unsigned)y < (unsigned)H && (unsigned)x < (unsigned)W)
      val = sb[((size_t)c * H + y) * W + x];
    xf[((size_t)b * L + l) * DQ + d] = val;
    sumsq += val * val;
  }
  for (int e = t; e < DV; e += 64) {
    const int c = e / 16, r = e % 16, u = r / 4, v = r % 4;
    const int y = 2 * hs + u - 1, x = 2 * ws + v - 1;
    float val = 0.f;
    if ((unsigned)y < (unsigned)H && (unsigned)x < (unsigned)W)
      val = sb[((size_t)c * H + y) * W + x];
    rw[((size_t)b * L + l) * DV + e] = val;
  }

  __shared__ float red[64];
  red[t] = sumsq;
  __syncthreads();
  if (t == 0) {
    float tot = 0.f;
    for (int i = 0; i < 64; ++i) tot += red[i];
    // mm[l]: mean of 3x3 patch of mask[::2,::2] (zero-padded) == 0 ? 1 : 0
    float msum = 0.f;
    for (int u = 0; u < 3; ++u)
      for (int v = 0; v < 3; ++v) {
        const int ys = hs + u - 1, xs = ws + v - 1;
        if ((unsigned)ys < (unsigned)HS && (unsigned)xs < (unsigned)HS)
          msum += mask[(size_t)(2 * ys) * W + 2 * xs];
      }
    const float mmv = ((msum / 9.0f) == 0.0f) ? 1.0f : 0.0f;
    scale[b * L + l] = 10.0f * mmv / fmaxf(sqrtf(tot), 1e-4f);
    if (b == 0) mm[l] = mmv;
  }
}

// ---------------------------------------------------------------------------
// 2) S[l,p] = scale[l] * G[l,p],  G = xf xf^T  (FP32 WMMA, symmetric).
//    Only tile blocks with J>=I are computed; each writes both the direct
//    tile (scale[l]*G) and the mirrored tile (scale[p]*G^T). Diagonal blocks
//    double-write identical values (benign).
//    block = 4 waves, each wave 32x32 as 2x2 tiles of 16x16, K-step 4.
// ---------------------------------------------------------------------------
__global__ void gemm1_kernel(const float* __restrict__ xf,
                             const float* __restrict__ scale,
                             float* __restrict__ S) {
  const int I = blockIdx.y, J = blockIdx.x;
  if (J < I) return;  // uniform exit before any WMMA
  const int lane = threadIdx.x & 31;
  const int wave = threadIdx.x >> 5;
  const int b = blockIdx.z;
  const int rowBase = I * 64 + (wave >> 1) * 32;  // l
  const int colBase = J * 64 + (wave & 1) * 32;   // p
  const int lm = lane & 15;
  const int kOff = (lane >> 4) * 2;  // 0 or 2 (K half per lane group)
  const float* xfb = xf + (size_t)b * L * DQ;
  const float* scb = scale + (size_t)b * L;

  v8f acc[2][2] = {};
  for (int k0 = 0; k0 < DQ; k0 += 4) {
    v2f a[2], bb[2];
#pragma unroll
    for (int i = 0; i < 2; ++i) {
      const int r = rowBase + i * 16 + lm;
      a[i] = *(const v2f*)(xfb + (size_t)r * DQ + k0 + kOff);
      const int c = colBase + i * 16 + lm;
      bb[i] = *(const v2f*)(xfb + (size_t)c * DQ + k0 + kOff);
    }
    // Unconditional WGP-scope prefetch (speculative; stays inside workspace).
    __builtin_prefetch(xfb + (size_t)(rowBase + lm) * DQ + k0 + 64, 0, 3);
    __builtin_prefetch(xfb + (size_t)(colBase + lm) * DQ + k0 + 64, 0, 3);
#pragma unroll
    for (int i = 0; i < 2; ++i)
#pragma unroll
      for (int j = 0; j < 2; ++j)
        acc[i][j] = __builtin_amdgcn_wmma_f32_16x16x4_f32(
            false, a[i], false, bb[j], (short)0, acc[i][j], false, false);
  }

  const int mHi = (lane >> 4) * 8;  // C/D layout: lanes16-31 hold M+8
  // direct tile: S[l,p] = scale[l]*G  (coalesced rows)
#pragma unroll
  for (int i = 0; i < 2; ++i) {
#pragma unroll
    for (int v = 0; v < 8; ++v) {
      const int lrow = rowBase + i * 16 + mHi + v;
      const float sc = scb[lrow];
      float* srow = S + ((size_t)b * L + lrow) * L;
#pragma unroll
      for (int j = 0; j < 2; ++j)
        srow[colBase + j * 16 + lm] = acc[i][j][v] * sc;
    }
  }
  // mirrored tile: S[p,l] = scale[p]*G[l,p]
#pragma unroll
  for (int i = 0; i < 2; ++i) {
#pragma unroll
    for (int j = 0; j < 2; ++j) {
      const int p = colBase + j * 16 + lm;
      const float scp = scb[p];
      float* scol = S + ((size_t)b * L + p) * L;
#pragma unroll
      for (int v = 0; v < 8; ++v) {
        const int lrow = rowBase + i * 16 + mHi + v;
        scol[lrow] = acc[i][j][v] * scp;
      }
    }
  }
}

// ---------------------------------------------------------------------------
// 3) Column softmax over l (axis=1), in place, times mm[l] afterwards.
//    One thread per column p; strided rows -> coalesced across threads.
//    41 MB score matrix is L2-resident (192 MB L2) so 3 passes are cheap.
// ---------------------------------------------------------------------------
__global__ void softmax_kernel(float* __restrict__ S,
                               const float* __restrict__ mm) {
  const int p = blockIdx.x * 256 + threadIdx.x;
  const int b = blockIdx.y;
  if (p >= L) return;
  float* col = S + (size_t)b * L * L + p;
  float mx = -3.402823466e38f;
  for (int l = 0; l < L; ++l) mx = fmaxf(mx, col[(size_t)l * L]);
  float s = 0.f;
  for (int l = 0; l < L; ++l) s += expf(col[(size_t)l * L] - mx);
  const float inv = 1.0f / s;
  for (int l = 0; l < L; ++l)
    col[(size_t)l * L] = mm[l] * expf(col[(size_t)l * L] - mx) * inv;
}

// ---------------------------------------------------------------------------
// 4) patches[p,d] = sum_l yi[l,p] * raw_w[l,d]  ->  C = yi^T * raw_w
//    A[m,k] = yi[k*L + m] (transposed gather), B[k,n] = raw_w[k*DV + n].
//    Wave tile 32x64 (2x4 WMMA tiles): 6 fragment loads : 8 WMMAs per K-step.
// ---------------------------------------------------------------------------
__global__ void gemm2_kernel(const float* __restrict__ yi,
                             const float* __restrict__ rw,
                             float* __restrict__ patches) {
  const int lane = threadIdx.x & 31;
  const int wave = threadIdx.x >> 5;
  const int b = blockIdx.z;
  const int rowBase = blockIdx.y * 64 + (wave >> 1) * 32;    // p (M=1600, 25)
  const int colBase = blockIdx.x * 128 + (wave & 1) * 64;    // d (N=1024, 8)
  const int lm = lane & 15;
  const int kOff = (lane >> 4) * 2;
  const float* yib = yi + (size_t)b * L * L;
  const float* rwb = rw + (size_t)b * L * DV;

  v8f acc[2][4] = {};
  for (int k0 = 0; k0 < L; k0 += 4) {
    const int k = k0 + kOff;
    v2f a[2], bb[4];
#pragma unroll
    for (int i = 0; i < 2; ++i) {
      const int m = rowBase + i * 16 + lm;
      a[i].x = yib[(size_t)k * L + m];
      a[i].y = yib[(size_t)(k + 1) * L + m];
    }
#pragma unroll
    for (int j = 0; j < 4; ++j) {
      const int n = colBase + j * 16 + lm;
      bb[j].x = rwb[(size_t)k * DV + n];
      bb[j].y = rwb[(size_t)(k + 1) * DV + n];
    }
    __builtin_prefetch(yib + (size_t)(k + 8) * L + rowBase + lm, 0, 3);
    __builtin_prefetch(rwb + (size_t)(k + 8) * DV + colBase + lm, 0, 3);
#pragma unroll
    for (int i = 0; i < 2; ++i)
#pragma unroll
      for (int j = 0; j < 4; ++j)
        acc[i][j] = __builtin_amdgcn_wmma_f32_16x16x4_f32(
            false, a[i], false, bb[j], (short)0, acc[i][j], false, false);
  }

  const int mHi = (lane >> 4) * 8;
#pragma unroll
  for (int i = 0; i < 2; ++i)
#pragma unroll
    for (int v = 0; v < 8; ++v) {
      const int p = rowBase + i * 16 + mHi + v;
      float* prow = patches + ((size_t)b * L + p) * DV;
#pragma unroll
      for (int j = 0; j < 4; ++j)
        prow[colBase + j * 16 + lm] = acc[i][j][v];
    }
}

// ---------------------------------------------------------------------------
// 5) Overlap-add reconstruction: canvas 82x82, rate 2, crop [1:81], /4.
//    Inverse gather: out[y,x] sums <=4 patch contributions.
// ---------------------------------------------------------------------------
__global__ void recon_kernel(const float* __restrict__ patches,
                             float* __restrict__ out) {
  const size_t idx = (size_t)blockIdx.x * 256 + threadIdx.x;
  const size_t total = (size_t)NB * NC * H * W;
  if (idx >= total) return;
  const int x = (int)(idx % W);
  const int y = (int)((idx / W) % H);
  const int c = (int)((idx / (H * W)) % NC);
  const int b = (int)(idx / ((size_t)NC * H * W));
  const int yc = y + 1, xc = x + 1;
  const float* pb = patches + (size_t)b * L * DV;
  float acc = 0.f;
#pragma unroll
  for (int du = 0; du < 2; ++du) {
    const int u = (yc & 1) + 2 * du;
    const int i = (yc - u) >> 1;
    if (i < 0 || i > HS - 1) continue;
#pragma unroll
    for (int dv = 0; dv < 2; ++dv) {
      const int v = (xc & 1) + 2 * dv;
      const int j = (xc - v) >> 1;
      if (j < 0 || j > HS - 1) continue;
      acc += pb[(size_t)(i * HS + j) * DV + c * 16 + u * 4 + v];
    }
  }
  out[idx] = acc * 0.25f;
}

// ---------------------------------------------------------------------------
extern "C" void kernel_launch(void* const* d_in, const int* in_sizes, int n_in,
                              void* d_out, int out_size, void* d_ws,
                              size_t ws_size, hipStream_t stream) {
  const float* bsrc = (const float*)d_in[0];   // (4,64,80,80)
  const float* mask = (const float*)d_in[1];   // (1,1,80,80)
  float* out = (float*)d_out;
  float* ws = (float*)d_ws;

  float* xf    = ws + XF_OFF;
  float* scale = ws + SCALE_OFF;
  float* mm    = ws + MM_OFF;
  float* rw    = ws + RW_OFF;
  float* S     = ws + S_OFF;
  float* pat   = ws + P_OFF;

  prep_kernel<<<dim3(L, NB), 64, 0, stream>>>(bsrc, mask, xf, rw, scale, mm);
  gemm1_kernel<<<dim3(L / 64, L / 64, NB), 128, 0, stream>>>(xf, scale, S);
  softmax_kernel<<<dim3((L + 255) / 256, NB), 256, 0, stream>>>(S, mm);
  gemm2_kernel<<<dim3(DV / 128, L / 64, NB), 128, 0, stream>>>(S, rw, pat);
  const size_t total = (size_t)NB * NC * H * W;
  recon_kernel<<<dim3((unsigned)((total + 255) / 256)), 256, 0, stream>>>(pat, out);
}